// HBVTorch_11106785427541
// MI455X (gfx1250) — compile-verified
//
#include <hip/hip_runtime.h>

// ---------------------------------------------------------------------------
// HBV hydrological model scan on gfx1250 (MI455X).
//   - Nonlinear per-basin recurrence over 2000 steps -> no parallel scan, no
//     WMMA applicability. 4096 basins = 128 wave32s => issue/latency bound.
//   - 1 lane = 1 basin; 32-thread single-wave workgroups spread across WGPs.
//   - CDNA5 async path: double-buffered LDS staging of forcing data with
//     global_load_async_to_lds_b128 (SADDR form, ASYNCcnt) + s_wait_asynccnt.
//   - Per-step math minimized: raw v_exp_f32/v_log_f32/v_rcp_f32/v_med3_f32.
// ---------------------------------------------------------------------------

#define LANES 32
#define CHUNK 16                                      // timesteps per buffer
#define ROW_BYTES (LANES * 12)                        // 384 B per timestep row
#define CHUNK_B128 ((CHUNK * ROW_BYTES) / (32 * 16))  // 12 async b128 / chunk

__constant__ float c_pmin[14] = {-2.5f, 0.5f, 0.4f, 0.0f, 0.0f, 50.0f, 0.3f,
                                  1.0f, 0.0f, 0.0f, 0.05f, 0.01f, 0.001f, 1.0f};
__constant__ float c_pmax[14] = { 2.5f, 10.0f, 1.5f, 0.1f, 0.2f, 700.0f, 1.0f,
                                  6.0f, 8.0f, 100.0f, 0.5f, 0.3f, 0.15f, 7.0f};

#define WAIT_ASYNCCNT(n) asm volatile("s_wait_asynccnt %0" :: "n"(n) : "memory")

#if __has_builtin(__builtin_amdgcn_exp2f)
#define EXP2F(x) __builtin_amdgcn_exp2f(x)     // raw v_exp_f32
#else
#define EXP2F(x) __builtin_exp2f(x)
#endif
#if __has_builtin(__builtin_amdgcn_logf)
#define LOG2F(x) __builtin_amdgcn_logf(x)      // raw v_log_f32
#else
#define LOG2F(x) __log2f(x)
#endif
#if __has_builtin(__builtin_amdgcn_fmed3f)
#define CLAMP0(x, hi) __builtin_amdgcn_fmed3f((x), 0.0f, (hi))
#else
#define CLAMP0(x, hi) fminf(fmaxf((x), 0.0f), (hi))
#endif

#define LOG2E 1.4426950408889634f

__device__ __forceinline__ float fast_sigmoid(float x) {
    return __builtin_amdgcn_rcpf(1.0f + EXP2F(-x * LOG2E));
}

__global__ __launch_bounds__(LANES) void hbv_scan_kernel(
    const float* __restrict__ x_phy,   // (n_t, n_b, 3)
    const float* __restrict__ params,  // (n_b, 14)
    float* __restrict__ q,             // (n_t, n_b)
    int n_t, int n_b)
{
    __shared__ float smem[2][CHUNK * LANES * 3];      // 2 x 6144 B

    const int lane = threadIdx.x;                     // 0..31
    const int b0   = blockIdx.x * LANES;              // wave's first basin
    const int b    = b0 + lane;

    // ---- per-basin parameter transform: p = pmin + sigmoid(w)*(pmax - pmin)
    float p[13];
#pragma unroll
    for (int i = 0; i < 13; ++i) {
        float w = params[(size_t)b * 14 + i];
        p[i] = c_pmin[i] + fast_sigmoid(w) * (c_pmax[i] - c_pmin[i]);
    }
    const float TT = p[0], CFMAX = p[1], SFCF = p[2], CFR = p[3], CWH = p[4];
    const float FC = p[5], LP = p[6], BETA = p[7], PERC = p[8], UZL = p[9];
    const float K0 = p[10], K1 = p[11], K2 = p[12];

    const float invFC     = __builtin_amdgcn_rcpf(FC);
    const float invFCLP   = __builtin_amdgcn_rcpf(FC * LP);
    const float cfr_cfmax = CFR * CFMAX;
    // sigmoid((TT-T)*5) = rcp(1 + exp2(T*c1 + c2))
    const float c1 = 5.0f * LOG2E;
    const float c2 = -TT * c1;

    float sp = 0.0f, lw = 0.0f, sm = FC * 0.5f, uz = 0.0f, lz = 0.0f;

    // ---- async-copy geometry -----------------------------------------------
    const unsigned row_stride  = (unsigned)n_b * 12u;          // bytes / row
    const unsigned long long chunk_stride = (unsigned long long)row_stride * CHUNK;
    // Wave's 384 B window inside each row (uniform across lanes).
    unsigned long long sbase =
        (unsigned long long)x_phy + (unsigned long long)b0 * 12u;

    // Per-lane segment offsets, computed ONCE: seg s = j*32+lane; 24 b128
    // segments per 384 B row; row = s/24, col = s%24.
    unsigned v_off[CHUNK_B128];   // global byte offset within a chunk
    unsigned l_off[CHUNK_B128];   // LDS byte offset within a buffer
#pragma unroll
    for (int j = 0; j < CHUNK_B128; ++j) {
        unsigned s   = (unsigned)(j * 32 + lane);
        unsigned row = s / 24u;
        unsigned col = s - row * 24u;
        v_off[j] = row * row_stride + col * 16u;
        l_off[j] = row * (unsigned)ROW_BYTES + col * 16u;
    }

    const unsigned lds_buf0 = (unsigned)(unsigned long long)(&smem[0][0]);
    const unsigned lds_buf1 = (unsigned)(unsigned long long)(&smem[1][0]);

    // Issue one chunk: 12 async b128 loads, SADDR form (scalar base + voffset).
    auto issue_chunk = [&](unsigned long long gbase, unsigned lbase) {
#pragma unroll
        for (int j = 0; j < CHUNK_B128; ++j) {
            unsigned laddr = lbase + l_off[j];
            asm volatile("global_load_async_to_lds_b128 %0, %1, %2"
                         :: "v"(laddr), "v"(v_off[j]), "s"(gbase)
                         : "memory");
        }
    };

    // ---- one HBV step (pure VALU; raw trans ops on the TRANS pipe) ---------
    auto step = [&](float P, float T, float E, int t) {
        float dT   = T - TT;
        float sig  = __builtin_amdgcn_rcpf(1.0f + EXP2F(fmaf(T, c1, c2)));
        float Psf  = P * sig;
        float rain = P - Psf;
        float snow = Psf * SFCF;
        float melt     = fminf(CFMAX * fmaxf(dT, 0.0f), sp);
        float refreeze = fminf(cfr_cfmax * fmaxf(-dT, 0.0f), lw);
        sp = sp + snow + refreeze - melt;
        lw = lw + melt - refreeze;
        float release = fmaxf(fmaf(-CWH, sp, lw), 0.0f);
        lw -= release;
        float wi  = rain + release;
        float aet = fminf(E * fminf(sm * invFCLP, 1.0f), sm);
        // pow(sm/FC, BETA): log2(0) = -inf -> exp2(-inf) = 0 == pow(0, BETA)
        float pw       = EXP2F(BETA * LOG2F(sm * invFC));
        float recharge = fminf(wi * pw, wi);
        sm = CLAMP0(sm + wi - aet - recharge, FC);
        float perc = fminf(PERC, uz);
        float fast = K0 * fmaxf(uz - UZL, 0.0f);
        float slow = K1 * uz;
        float base = K2 * lz;
        float fs   = fast + slow;
        __builtin_nontemporal_store(fs + base, &q[(size_t)t * n_b + b]);
        uz = fmaxf(uz + recharge - perc - fs, 0.0f);
        lz = fmaxf(lz + perc - base, 0.0f);
    };

    const int nchunks = n_t / CHUNK;

    if (nchunks > 0) {
        issue_chunk(sbase, lds_buf0);
        unsigned long long snext = sbase + chunk_stride;
        for (int c = 0; c < nchunks; ++c) {
            if (c + 1 < nchunks) {
                // Prefetch next chunk into the other buffer; async loads
                // complete in order, so asynccnt <= 12 => current chunk ready.
                issue_chunk(snext, ((c + 1) & 1) ? lds_buf1 : lds_buf0);
                snext += chunk_stride;
                WAIT_ASYNCCNT(CHUNK_B128);
            } else {
                WAIT_ASYNCCNT(0);
            }
            const float* buf = (c & 1) ? &smem[1][0] : &smem[0][0];
#pragma unroll
            for (int tl = 0; tl < CHUNK; ++tl) {
                float P = buf[tl * (LANES * 3) + lane * 3 + 0];
                float T = buf[tl * (LANES * 3) + lane * 3 + 1];
                float E = buf[tl * (LANES * 3) + lane * 3 + 2];
                step(P, T, E, c * CHUNK + tl);
            }
        }
    }

    // Remainder timesteps (none for n_t=2000): straight global loads.
    for (int t = nchunks * CHUNK; t < n_t; ++t) {
        const float* xp = x_phy + ((size_t)t * n_b + b) * 3;
        step(xp[0], xp[1], xp[2], t);
    }
}

extern "C" void kernel_launch(void* const* d_in, const int* in_sizes, int n_in,
                              void* d_out, int out_size, void* d_ws, size_t ws_size,
                              hipStream_t stream) {
    const float* x_phy  = (const float*)d_in[0];
    const float* params = (const float*)d_in[1];
    float* out = (float*)d_out;

    const int n_b = in_sizes[1] / 14;            // 4096
    const int n_t = in_sizes[0] / (n_b * 3);     // 2000

    const int nblocks = n_b / LANES;             // 128 single-wave workgroups
    hbv_scan_kernel<<<dim3(nblocks), dim3(LANES), 0, stream>>>(
        x_phy, params, out, n_t, n_b);
}